// XMLModel_54073638256944
// MI455X (gfx1250) — compile-verified
//
#include <hip/hip_runtime.h>
#include <math.h>
#include <stdint.h>

// ---------------------------------------------------------------------------
// Problem constants (from reference)
// ---------------------------------------------------------------------------
#define B_DIM   256
#define IN_DIM  100000
#define HID     512
#define ODIM    30000
#define S_CNT   256      // codebook groups
#define KS_CNT  256      // codes per codebook
#define NCHUNK  25       // split-K chunks for GEMM1
#define KCHUNK  4000     // 25 * 4000 = 100000

typedef float v2f __attribute__((ext_vector_type(2)));
typedef float v8f __attribute__((ext_vector_type(8)));
typedef int   v4i __attribute__((ext_vector_type(4)));

// ---------------------------------------------------------------------------
// CDNA5 async global->LDS copy (GLOBAL_LOAD_ASYNC_TO_LDS_B128, ASYNCcnt).
// Builtin signature (from hipcc diagnostic): (v4i AS1*, v4i AS3*, Imm, Imm).
// Guarded: falls back to VGPR-staged b128 copies if the builtin is absent.
// ---------------------------------------------------------------------------
#if __has_builtin(__builtin_amdgcn_global_load_async_to_lds_b128)
#define HAS_ASYNC_LDS 1
#else
#define HAS_ASYNC_LDS 0
#endif

#if HAS_ASYNC_LDS
typedef __attribute__((address_space(1))) v4i* gv4i_p;
typedef __attribute__((address_space(3))) v4i* lv4i_p;

__device__ __forceinline__ void async_cp_b128(float* lds_dst, const float* g_src) {
    __builtin_amdgcn_global_load_async_to_lds_b128(
        (gv4i_p)(uintptr_t)g_src,
        (lv4i_p)(uintptr_t)lds_dst,
        /*offset=*/0, /*cpol=*/0);
}
__device__ __forceinline__ void wait_async0() {
    asm volatile("s_wait_asynccnt 0" ::: "memory");
}
#endif

// ---------------------------------------------------------------------------
// k0: zero the h accumulator (must re-run every launch: atomics accumulate)
// ---------------------------------------------------------------------------
__global__ void zero_ws(float* __restrict__ p, int n) {
    int i = blockIdx.x * blockDim.x + threadIdx.x;
    if (i < n) p[i] = 0.0f;
}

// ---------------------------------------------------------------------------
// k1: split-K GEMM1.  h_acc[b,h] += sum_k x[b,k] * w1[h,k]
// Workgroup: 64(B) x 64(H) tile over one K chunk of 4000.
// 16 waves, each owns one 16x16 WMMA tile (4x4 wave grid).
// Tiles staged via async global->LDS DMA (VGPR-free) when available.
// ---------------------------------------------------------------------------
#define LPAD 68   // 64 + 4 pad words -> conflict-free A/B fragment reads

__global__ void __launch_bounds__(512)
gemm1_splitk(const float* __restrict__ x, const float* __restrict__ w1,
             float* __restrict__ hacc)
{
    __shared__ float xs[64 * LPAD];
    __shared__ float ws[64 * LPAD];

    const int tid   = threadIdx.x;
    const int lane  = tid & 31;
    const int wave  = tid >> 5;
    const int mw    = wave & 3;         // M sub-tile (x rows)
    const int nw    = wave >> 2;        // N sub-tile (w1 rows = h cols)
    const int b0    = (blockIdx.x & 3) * 64;    // 256/64 = 4 B tiles
    const int h0    = (blockIdx.x >> 2) * 64;   // 512/64 = 8 H tiles
    const int kbase = blockIdx.y * KCHUNK;
    const int kend  = kbase + KCHUNK;

    const int mLane = lane & 15;
    const int kSel  = (lane >> 4) << 1;  // 0 or 2: K-pair select (A/B frag layout)

    // cooperative tile-load mapping: 512 threads * 8 floats = 64x64
    const int lr = tid >> 3;            // 0..63 tile row
    const int lc = (tid & 7) * 8;       // col start (8 consecutive floats)

    const float* xg0 = &x[(b0 + lr) * IN_DIM + lc];
    const float* wg0 = &w1[(h0 + lr) * IN_DIM + lc];
    float* xl0 = &xs[lr * LPAD + lc];
    float* wl0 = &ws[lr * LPAD + lc];

    v8f acc = {};

    for (int kb = kbase; kb < kend; kb += 64) {
        if (kb + 64 <= kend) {
#if HAS_ASYNC_LDS
            // CDNA5 path: DMA straight into LDS, no VGPR staging
            async_cp_b128(xl0,     xg0 + kb);
            async_cp_b128(xl0 + 4, xg0 + kb + 4);
            async_cp_b128(wl0,     wg0 + kb);
            async_cp_b128(wl0 + 4, wg0 + kb + 4);
            if (kb + 128 <= kend) {   // prefetch next K block of the big streams
                __builtin_prefetch(xg0 + kb + 64, 0, 0);
                __builtin_prefetch(wg0 + kb + 64, 0, 0);
            }
            wait_async0();
#else
            const float4* xg = reinterpret_cast<const float4*>(xg0 + kb);
            const float4* wg = reinterpret_cast<const float4*>(wg0 + kb);
            float4 x0 = xg[0], x1 = xg[1];
            float4 w0 = wg[0], w1v = wg[1];
            *reinterpret_cast<float4*>(xl0)     = x0;
            *reinterpret_cast<float4*>(xl0 + 4) = x1;
            *reinterpret_cast<float4*>(wl0)     = w0;
            *reinterpret_cast<float4*>(wl0 + 4) = w1v;
#endif
        } else {
            // tail of the chunk (4000 % 64 == 32): zero-fill out-of-range K
            #pragma unroll
            for (int i = 0; i < 8; ++i) {
                int c = lc + i;
                bool ok = (kb + c) < kend;
                xl0[i] = ok ? xg0[kb + i] : 0.0f;
                wl0[i] = ok ? wg0[kb + i] : 0.0f;
            }
        }
        __syncthreads();

        const float* xrow = &xs[(mw * 16 + mLane) * LPAD + kSel];
        const float* wrow = &ws[(nw * 16 + mLane) * LPAD + kSel];
        #pragma unroll
        for (int kk = 0; kk < 64; kk += 4) {
            v2f a, b;
            a.x = xrow[kk];  a.y = xrow[kk + 1];
            b.x = wrow[kk];  b.y = wrow[kk + 1];
            acc = __builtin_amdgcn_wmma_f32_16x16x4_f32(
                false, a, false, b, (short)0, acc, false, false);
        }
        __syncthreads();
    }

    // scatter-accumulate the 16x16 C fragment (split-K reduction via L2 atomics)
    const int row0 = b0 + mw * 16 + ((lane >> 4) << 3);  // +8 for upper half-wave
    const int col  = h0 + nw * 16 + mLane;
    #pragma unroll
    for (int j = 0; j < 8; ++j) {
        unsafeAtomicAdd(&hacc[(row0 + j) * HID + col], acc[j]);
    }
}

// ---------------------------------------------------------------------------
// k2: h = relu(h_acc + b1)   (in place in workspace)
// ---------------------------------------------------------------------------
__global__ void bias_relu(float* __restrict__ h, const float* __restrict__ b1) {
    int i = blockIdx.x * blockDim.x + threadIdx.x;
    if (i < B_DIM * HID) {
        float v = h[i] + b1[i & (HID - 1)];
        h[i] = v > 0.0f ? v : 0.0f;
    }
}

// ---------------------------------------------------------------------------
// k3: fused  PQ-decode(w2 tile) -> GEMM2 -> +b2 -> log_softmax(axis=0) -> out
// One workgroup = 16 output columns x all 256 batch rows, so the axis-0
// (over-batch) softmax reduction never leaves LDS.
// 16 waves: wave w computes batch rows [16w, 16w+16) with K=512 WMMA loop.
// ---------------------------------------------------------------------------
#define WPAD 17   // 16 + 1 pad for LDS banking

__global__ void __launch_bounds__(512)
gemm2_fused(const float* __restrict__ h, const float* __restrict__ cb,
            const int* __restrict__ codes, const float* __restrict__ b2,
            float* __restrict__ out)
{
    __shared__ float w2t[HID * WPAD];      // decoded w2 tile, [k][n] padded (34 KB)
    __shared__ float lt[B_DIM * WPAD];     // logits tile, [row][n] padded (17 KB)
    __shared__ float pm[32 * 16];          // partial reduce
    __shared__ float cmax[16];
    __shared__ float clog[16];

    const int tid  = threadIdx.x;
    const int lane = tid & 31;
    const int wave = tid >> 5;
    const int o0   = blockIdx.x * 16;      // 30000/16 = 1875 (exact)

    // ---- decode 16 columns of w2 from codebooks into LDS -------------------
    // 16 cols * 256 groups = 4096 code lookups, 8 per thread, coalesced codes
    #pragma unroll
    for (int i = 0; i < 8; ++i) {
        int e = i * 512 + tid;
        int o = e >> 8;                    // 0..15 column within tile
        int s = e & 255;                   // codebook group
        int c = codes[(o0 + o) * S_CNT + s];
        const float* p = &cb[((s << 8) + c) * 2];
        w2t[(2 * s)     * WPAD + o] = p[0];
        w2t[(2 * s + 1) * WPAD + o] = p[1];
    }
    __syncthreads();

    // ---- GEMM2: 16x16 tile per wave, K = 512 -------------------------------
    const int mLane = lane & 15;
    const int kSel  = (lane >> 4) << 1;
    const int m0    = wave * 16;

    v8f acc = {};
    const float* hrow = &h[(m0 + mLane) * HID + kSel];
    #pragma unroll 8
    for (int k = 0; k < HID; k += 4) {
        v2f a, b;
        a.x = hrow[k];  a.y = hrow[k + 1];
        b.x = w2t[(k + kSel)     * WPAD + mLane];
        b.y = w2t[(k + kSel + 1) * WPAD + mLane];
        acc = __builtin_amdgcn_wmma_f32_16x16x4_f32(
            false, a, false, b, (short)0, acc, false, false);
    }

    // deposit logits (+b2; constant per column, kept for fidelity) into LDS
    const float bv   = b2[o0 + mLane];
    const int   rsel = (lane >> 4) << 3;
    #pragma unroll
    for (int j = 0; j < 8; ++j) {
        lt[(m0 + rsel + j) * WPAD + mLane] = acc[j] + bv;
    }
    __syncthreads();

    // ---- log_softmax over axis=0 (the 256 rows), per column ----------------
    const int col = tid & 15;
    const int g   = tid >> 4;              // 0..31 row-group, 8 rows each
    float mx = -INFINITY;
    #pragma unroll
    for (int i = 0; i < 8; ++i) mx = fmaxf(mx, lt[(g * 8 + i) * WPAD + col]);
    pm[g * 16 + col] = mx;
    __syncthreads();
    if (tid < 16) {
        float m2 = pm[col];
        for (int gg = 1; gg < 32; ++gg) m2 = fmaxf(m2, pm[gg * 16 + col]);
        cmax[col] = m2;
    }
    __syncthreads();
    const float cm = cmax[col];
    float sm = 0.0f;
    #pragma unroll
    for (int i = 0; i < 8; ++i) sm += __expf(lt[(g * 8 + i) * WPAD + col] - cm);
    pm[g * 16 + col] = sm;
    __syncthreads();
    if (tid < 16) {
        float s2 = 0.0f;
        for (int gg = 0; gg < 32; ++gg) s2 += pm[gg * 16 + col];
        clog[col] = __logf(s2);
    }
    __syncthreads();
    const float cl = clog[col];
    #pragma unroll
    for (int i = 0; i < 8; ++i) {
        int r = g * 8 + i;
        out[r * ODIM + o0 + col] = lt[r * WPAD + col] - cm - cl;
    }
}

// ---------------------------------------------------------------------------
// launcher
// ---------------------------------------------------------------------------
extern "C" void kernel_launch(void* const* d_in, const int* in_sizes, int n_in,
                              void* d_out, int out_size, void* d_ws, size_t ws_size,
                              hipStream_t stream) {
    const float* x     = (const float*)d_in[0];
    const float* w1    = (const float*)d_in[1];
    const float* b1    = (const float*)d_in[2];
    const float* cb    = (const float*)d_in[3];
    const int*   codes = (const int*)d_in[4];
    const float* b2    = (const float*)d_in[5];
    float* out  = (float*)d_out;
    float* hacc = (float*)d_ws;   // 256*512 f32 = 512 KB

    (void)in_sizes; (void)n_in; (void)out_size; (void)ws_size;

    zero_ws<<<(B_DIM * HID + 255) / 256, 256, 0, stream>>>(hacc, B_DIM * HID);
    gemm1_splitk<<<dim3(32, NCHUNK), 512, 0, stream>>>(x, w1, hacc);
    bias_relu<<<(B_DIM * HID + 255) / 256, 256, 0, stream>>>(hacc, b1);
    gemm2_fused<<<ODIM / 16, 512, 0, stream>>>(hacc, cb, codes, b2, out);
}